// KeGNN_1571958030448
// MI455X (gfx1250) — compile-verified
//
#include <hip/hip_runtime.h>
#include <math.h>

typedef __attribute__((ext_vector_type(2))) float v2f;
typedef __attribute__((ext_vector_type(8))) float v8f;

// ---------------------------------------------------------------- utilities

__global__ void k_fill1(float* __restrict__ p, int n) {
  int i = blockIdx.x * blockDim.x + threadIdx.x;
  if (i < n) p[i] = 1.0f;
}

__global__ void k_degree(const int* __restrict__ dst, float* __restrict__ deg, int E) {
  int i = blockIdx.x * blockDim.x + threadIdx.x;
  if (i < E) atomicAdd(&deg[dst[i]], 1.0f);
}

__global__ void k_rsqrt_inplace(float* __restrict__ p, int n) {
  int i = blockIdx.x * blockDim.x + threadIdx.x;
  if (i < n) p[i] = rsqrtf(p[i]);
}

__global__ void k_edgenorm(const int* __restrict__ src, const int* __restrict__ dst,
                           const float* __restrict__ dinv, float* __restrict__ w, int E) {
  int i = blockIdx.x * blockDim.x + threadIdx.x;
  if (i < E) w[i] = dinv[src[i]] * dinv[dst[i]];
}

// self-loop contribution + bias:  agg[i,f] = h[i,f]*dinv[i]^2 + bias[f]
__global__ void k_selfinit(const float* __restrict__ h, const float* __restrict__ dinv,
                           const float* __restrict__ bias, float* __restrict__ agg,
                           int N, int F) {
  int i = blockIdx.x * blockDim.x + threadIdx.x;
  if (i >= N * F) return;
  int node = i / F;
  int f    = i - node * F;
  float di = dinv[node];
  agg[i] = h[i] * di * di + bias[f];
}

// wave-per-edge scatter-add: agg[dst] += h[src] * w[e]  (F/4 lanes x float4)
__global__ void k_scatter(const float* __restrict__ h, const float* __restrict__ wn,
                          const int* __restrict__ src, const int* __restrict__ dst,
                          float* __restrict__ agg, int E, int shift, int F) {
  int t = blockIdx.x * blockDim.x + threadIdx.x;
  int e = t >> shift;
  if (e >= E) return;
  int f4 = (t & ((1 << shift) - 1)) << 2;
  int s = src[e], d = dst[e];
  float w = wn[e];
  const float4 v = *(const float4*)(h + (size_t)s * F + f4);
  float* op = agg + (size_t)d * F + f4;
  atomicAdd(op + 0, v.x * w);
  atomicAdd(op + 1, v.y * w);
  atomicAdd(op + 2, v.z * w);
  atomicAdd(op + 3, v.w * w);
}

// -------------------------------------------------- B repack into WMMA order
// Source: row-major B[K][Nc]. Dest (v2f units): for each k-step kk (=k0/4),
// khalf in {0,1}, column n: pair { B[kk*4+khalf*2+0][n], B[kk*4+khalf*2+1][n] }
// at v2f index ((kk*2 + khalf)*Nc + n). Lane n reads one contiguous b64.
__global__ void k_packB(const float* __restrict__ B, float* __restrict__ Bp, int K, int Nc) {
  int i = blockIdx.x * blockDim.x + threadIdx.x;
  if (i >= K * Nc) return;
  int v    = i & 1;
  int r    = i >> 1;
  int n    = r % Nc;
  int g    = r / Nc;      // g = kk*2 + khalf
  int half = g & 1;
  int kk   = g >> 1;
  Bp[i] = B[(size_t)(kk * 4 + half * 2 + v) * Nc + n];
}

// ----------------------------------------------------- fp32 WMMA GEMM (exact)
// C[N x (NT*16)] = A[N x K] * B[K x (NT*16)], one wave per 16-row strip.
// V_WMMA_F32_16X16X4_F32 lane layouts (ISA 7.12.2, 32-bit tables):
//   A: lane L holds A[M=L%16][k0 + 2*(L/16) + v]  in VGPR v (v=0,1)
//   B: lane L holds B[k0 + 2*(L/16) + v][N=L%16]  in VGPR v
//   C: lane L, VGPR r holds C[M = r + 8*(L/16)][N = L%16]
template <int NT, bool RELU>
__global__ void k_gemm_wmma(const float* __restrict__ A, const float* __restrict__ Bp,
                            float* __restrict__ C, int K, int N, int rowTiles) {
  const int wid = (blockIdx.x * blockDim.x + threadIdx.x) >> 5;
  if (wid >= rowTiles) return;                 // wave-uniform exit; EXEC stays all-1s
  const int lane  = threadIdx.x & 31;
  const int mrow  = lane & 15;
  const int khalf = lane >> 4;
  const int Nc    = NT * 16;

  v8f acc[NT] = {};
  int arowi = wid * 16 + mrow;
  if (arowi >= N) arowi = N - 1;               // safe clamp (N%16==0 here anyway)
  const float* arow = A + (size_t)arowi * K;
  const v2f* bpk = (const v2f*)Bp + (size_t)khalf * Nc + mrow;

#pragma unroll 2
  for (int k0 = 0; k0 < K; k0 += 4) {
    v2f a = *(const v2f*)(arow + k0 + 2 * khalf);
    if (RELU) { a.x = fmaxf(a.x, 0.0f); a.y = fmaxf(a.y, 0.0f); }
    const v2f* bp = bpk + (size_t)(k0 >> 2) * 2 * Nc;
    v2f b[NT];
#pragma unroll
    for (int t = 0; t < NT; ++t) b[t] = bp[t * 16];
#pragma unroll
    for (int t = 0; t < NT; ++t) {
      acc[t] = __builtin_amdgcn_wmma_f32_16x16x4_f32(
          /*neg_a=*/false, a, /*neg_b=*/false, b[t],
          /*c_mod=*/(short)0, acc[t], /*reuse_a=*/false, /*reuse_b=*/false);
    }
  }

#pragma unroll
  for (int t = 0; t < NT; ++t) {
#pragma unroll
    for (int r = 0; r < 8; ++r) {
      const int row = wid * 16 + r + 8 * khalf;
      if (row < N) C[(size_t)row * Nc + t * 16 + mrow] = acc[t][r];
    }
  }
}

// --------------------------------------------- per-node epilogue (wave/node)
// rule_out = h@We + be ; gate = sigmoid(rule_out@Wg + bg) ; h += gate*rule_w ;
// out = log_softmax(h)   with h of length 64 (2 feats per lane, wave32).
__global__ void k_epilogue(const float* __restrict__ agg2, const float* __restrict__ We,
                           const float* __restrict__ be, const float* __restrict__ Wg,
                           const float* __restrict__ bg, const float* __restrict__ rw,
                           float* __restrict__ out, int N) {
  const int wave = (blockIdx.x * blockDim.x + threadIdx.x) >> 5;
  const int lane = threadIdx.x & 31;
  if (wave >= N) return;
  const float* h = agg2 + (size_t)wave * 64;
  float h0 = h[lane];
  float h1 = h[lane + 32];

  float r0 = h0 * We[lane * 3 + 0] + h1 * We[(lane + 32) * 3 + 0];
  float r1 = h0 * We[lane * 3 + 1] + h1 * We[(lane + 32) * 3 + 1];
  float r2 = h0 * We[lane * 3 + 2] + h1 * We[(lane + 32) * 3 + 2];
#pragma unroll
  for (int m = 16; m >= 1; m >>= 1) {
    r0 += __shfl_xor(r0, m, 32);
    r1 += __shfl_xor(r1, m, 32);
    r2 += __shfl_xor(r2, m, 32);
  }
  r0 += be[0]; r1 += be[1]; r2 += be[2];
  const float z    = r0 * Wg[0] + r1 * Wg[1] + r2 * Wg[2] + bg[0];
  const float gate = 1.0f / (1.0f + expf(-z));
  const float add  = gate * rw[0];
  h0 += add; h1 += add;

  float mx = fmaxf(h0, h1);
#pragma unroll
  for (int m = 16; m >= 1; m >>= 1) mx = fmaxf(mx, __shfl_xor(mx, m, 32));
  float s = expf(h0 - mx) + expf(h1 - mx);
#pragma unroll
  for (int m = 16; m >= 1; m >>= 1) s += __shfl_xor(s, m, 32);
  const float lse = mx + logf(s);

  out[(size_t)wave * 64 + lane]      = h0 - lse;
  out[(size_t)wave * 64 + lane + 32] = h1 - lse;
}

// ------------------------------------------------------------------- driver

static inline int cdiv(long long a, long long b) { return (int)((a + b - 1) / b); }

extern "C" void kernel_launch(void* const* d_in, const int* in_sizes, int n_in,
                              void* d_out, int out_size, void* d_ws, size_t ws_size,
                              hipStream_t stream) {
  const float* x  = (const float*)d_in[0];
  const int*   ei = (const int*)d_in[1];
  // d_in[2] = rules: unused by the reference math
  const float* W1 = (const float*)d_in[3];
  const float* b1 = (const float*)d_in[4];
  const float* W2 = (const float*)d_in[5];
  const float* b2 = (const float*)d_in[6];
  const float* We = (const float*)d_in[7];
  const float* be = (const float*)d_in[8];
  const float* Wg = (const float*)d_in[9];
  const float* bg = (const float*)d_in[10];
  const float* rw = (const float*)d_in[11];
  float* out = (float*)d_out;

  const int IN  = 256, HID = 128, OUT = 64;
  const int N = in_sizes[0] / IN;  // 50000
  const int E = in_sizes[1] / 2;   // 800000
  const int* src = ei;
  const int* dst = ei + E;

  // workspace carve-up (floats)
  float* ws = (float*)d_ws;
  size_t o = 0;
  float* dinv  = ws + o; o += (size_t)((N + 63) / 64) * 64;
  float* wnorm = ws + o; o += (size_t)((E + 63) / 64) * 64;
  float* Bp1   = ws + o; o += (size_t)IN * HID;     // repacked W1
  float* Bp2   = ws + o; o += (size_t)HID * OUT;    // repacked W2
  float* h1    = ws + o; o += (size_t)N * HID;
  float* agg1  = ws + o; o += (size_t)N * HID;
  float* h2    = ws + o; o += (size_t)N * OUT;
  float* agg2  = ws + o;

  const int T = 256;

  // 1) degree (self-loop included via init=1.0) -> dinv -> per-edge norm
  k_fill1<<<cdiv(N, T), T, 0, stream>>>(dinv, N);
  k_degree<<<cdiv(E, T), T, 0, stream>>>(dst, dinv, E);
  k_rsqrt_inplace<<<cdiv(N, T), T, 0, stream>>>(dinv, N);
  k_edgenorm<<<cdiv(E, T), T, 0, stream>>>(src, dst, dinv, wnorm, E);

  // 2) repack weights into WMMA lane order (once per launch, tiny)
  k_packB<<<cdiv(IN * HID, T), T, 0, stream>>>(W1, Bp1, IN, HID);
  k_packB<<<cdiv(HID * OUT, T), T, 0, stream>>>(W2, Bp2, HID, OUT);

  // 3) layer 1: h1 = x @ W1 (WMMA f32), agg1 = selfloop+b1 + scatter
  const int rowTiles = (N + 15) / 16;  // 3125
  k_gemm_wmma<8, false><<<cdiv(rowTiles, 4), 128, 0, stream>>>(x, Bp1, h1, IN, N, rowTiles);
  k_selfinit<<<cdiv((long long)N * HID, T), T, 0, stream>>>(h1, dinv, b1, agg1, N, HID);
  k_scatter<<<cdiv((long long)E * (HID / 4), T), T, 0, stream>>>(h1, wnorm, src, dst, agg1,
                                                                 E, /*shift=*/5, HID);

  // 4) layer 2: h2 = relu(agg1) @ W2 (ReLU folded into A-load), agg2 likewise
  k_gemm_wmma<4, true><<<cdiv(rowTiles, 4), 128, 0, stream>>>(agg1, Bp2, h2, HID, N, rowTiles);
  k_selfinit<<<cdiv((long long)N * OUT, T), T, 0, stream>>>(h2, dinv, b2, agg2, N, OUT);
  k_scatter<<<cdiv((long long)E * (OUT / 4), T), T, 0, stream>>>(h2, wnorm, src, dst, agg2,
                                                                 E, /*shift=*/4, OUT);

  // 5) rule gate + log_softmax, wave per node
  k_epilogue<<<cdiv((long long)N * 32, T), T, 0, stream>>>(agg2, We, be, Wg, bg, rw, out, N);
}